// GlobalGNN_34703335752374
// MI455X (gfx1250) — compile-verified
//
#include <hip/hip_runtime.h>

#define D 128
#define EPS 1e-5f

typedef float v2f __attribute__((ext_vector_type(2)));
typedef float v8f __attribute__((ext_vector_type(8)));

// ---------------- degree / normalization ----------------
__global__ __launch_bounds__(256) void k_deg_init(float* deg, int n) {
  int i = blockIdx.x * 256 + threadIdx.x;
  if (i < n) deg[i] = 1.0f;  // self loop
}

__global__ __launch_bounds__(256) void k_deg_accum(float* deg, const int* dst, int e) {
  int i = blockIdx.x * 256 + threadIdx.x;
  if (i < e) atomicAdd(&deg[dst[i]], 1.0f);
}

__global__ __launch_bounds__(256) void k_dinv(float* deg, int n) {
  int i = blockIdx.x * 256 + threadIdx.x;
  if (i < n) deg[i] = __frsqrt_rn(deg[i]);  // deg >= 1 always
}

// ---------------- dense GEMM via V_WMMA_F32_16X16X4_F32 ----------------
// M = H @ W ; H: nrows x 128 row-major, W: 128 x 128 row-major.
// One wave per 16-row block; W staged in LDS (64 KB of the 320 KB WGP LDS).
__global__ __launch_bounds__(256) void k_gemm_wmma(const float* __restrict__ H,
                                                   const float* __restrict__ Wl,
                                                   float* __restrict__ M, int nrows) {
  __shared__ float sW[D * D];
  for (int i = threadIdx.x; i < D * D / 4; i += 256)
    ((float4*)sW)[i] = ((const float4*)Wl)[i];
  __syncthreads();

  const int wave   = threadIdx.x >> 5;
  const int lane   = threadIdx.x & 31;
  const int rowblk = blockIdx.x * 8 + wave;
  if (rowblk * 16 >= nrows) return;  // wave-uniform: EXEC stays all-1 for WMMA

  const int half  = lane >> 4;   // 0: K even pair, 1: K odd pair
  const int l15   = lane & 15;
  const int khalf = half * 2;
  const float* hrow = H + (size_t)(rowblk * 16 + l15) * D;

  v8f acc[8];
#pragma unroll
  for (int j = 0; j < 8; ++j) acc[j] = {};

  for (int k0 = 0; k0 < D; k0 += 4) {
    // A 16x4 f32 layout: lanes 0-15 -> K = k0+0,1 ; lanes 16-31 -> K = k0+2,3
    v2f a;
    a.x = hrow[k0 + khalf];
    a.y = hrow[k0 + khalf + 1];
#pragma unroll
    for (int j = 0; j < 8; ++j) {
      // B 4x16 f32 layout: col = lane&15, same K split across lane halves
      v2f bm;
      bm.x = sW[(k0 + khalf) * D + j * 16 + l15];
      bm.y = sW[(k0 + khalf + 1) * D + j * 16 + l15];
      acc[j] = __builtin_amdgcn_wmma_f32_16x16x4_f32(
          /*neg_a=*/false, a, /*neg_b=*/false, bm,
          /*c_mod=*/(short)0, acc[j], /*reuse_a=*/false, /*reuse_b=*/false);
    }
  }

  // C/D 16x16 f32 layout: VGPR v, lane L -> row = v + 8*(L>>4), col = L&15
  const int orow0 = rowblk * 16 + half * 8;
#pragma unroll
  for (int j = 0; j < 8; ++j) {
    float* op = M + (size_t)orow0 * D + j * 16 + l15;
#pragma unroll
    for (int v = 0; v < 8; ++v) op[(size_t)v * D] = acc[j][v];
  }
}

// ---------------- aggregation ----------------
// AGG[n,:] = M[n,:] * dinv[n]^2   (self-loop term; also serves as the zero-init)
__global__ __launch_bounds__(256) void k_agg_init(const float* __restrict__ M,
                                                  float* __restrict__ AGG,
                                                  const float* __restrict__ dinv, int n) {
  int i = blockIdx.x * 256 + threadIdx.x;  // one float4 per thread
  if (i >= n * 32) return;
  int node = i >> 5;
  float w = dinv[node];
  w = w * w;
  float4 m = ((const float4*)M)[i];
  float4 r = make_float4(m.x * w, m.y * w, m.z * w, m.w * w);
  ((float4*)AGG)[i] = r;
}

// One wave per edge: 32 lanes x float4 = 128 features, 4 f32 atomics per lane.
__global__ __launch_bounds__(256) void k_agg_edges(const float* __restrict__ M,
                                                   float* __restrict__ AGG,
                                                   const float* __restrict__ dinv,
                                                   const int* __restrict__ src,
                                                   const int* __restrict__ dst, int e) {
  int wv = (blockIdx.x * 256 + threadIdx.x) >> 5;
  int lane = threadIdx.x & 31;
  if (wv >= e) return;
  int s = src[wv], d = dst[wv];
  float nrm = dinv[s] * dinv[d];
  float4 mv = ((const float4*)(M + (size_t)s * D))[lane];
  float* op = AGG + (size_t)d * D + lane * 4;
  atomicAdd(op + 0, mv.x * nrm);
  atomicAdd(op + 1, mv.y * nrm);
  atomicAdd(op + 2, mv.z * nrm);
  atomicAdd(op + 3, mv.w * nrm);
}

// ---------------- bias + BN (running stats) + ReLU, in place ----------------
__global__ __launch_bounds__(256) void k_bn_relu(const float* __restrict__ AGG,
                                                 float* __restrict__ H,
                                                 const float* __restrict__ b,
                                                 const float* __restrict__ gamma,
                                                 const float* __restrict__ beta,
                                                 const float* __restrict__ rmean,
                                                 const float* __restrict__ rvar, int total) {
  int i = blockIdx.x * 256 + threadIdx.x;
  if (i >= total) return;
  int f = i & (D - 1);
  float v = AGG[i] + b[f];
  v = (v - rmean[f]) * __frsqrt_rn(rvar[f] + EPS) * gamma[f] + beta[f];
  H[i] = v > 0.f ? v : 0.f;
}

// ---------------- pooling ----------------
__global__ __launch_bounds__(256) void k_pool_zero(float* gsum, float* gmax, float* gcnt,
                                                   int g) {
  int i = blockIdx.x * 256 + threadIdx.x;
  if (i < g * D) { gsum[i] = 0.f; gmax[i] = 0.f; }
  if (i < g) gcnt[i] = 0.f;
}

__global__ __launch_bounds__(256) void k_pool_accum(const float* __restrict__ H,
                                                    const int* __restrict__ batch,
                                                    float* gsum, float* gmax, float* gcnt,
                                                    int n) {
  int i = blockIdx.x * 256 + threadIdx.x;  // one float4 per thread
  if (i >= n * 32) return;
  int node = i >> 5;
  int q = i & 31;
  int g = batch[node];
  float4 h4 = ((const float4*)(H + (size_t)node * D))[q];
  float* sp = gsum + (size_t)g * D + q * 4;
  atomicAdd(sp + 0, h4.x);
  atomicAdd(sp + 1, h4.y);
  atomicAdd(sp + 2, h4.z);
  atomicAdd(sp + 3, h4.w);
  // post-ReLU h >= 0, so int-ordered atomicMax matches float ordering
  int* mp = (int*)(gmax + (size_t)g * D + q * 4);
  atomicMax(mp + 0, __float_as_int(h4.x));
  atomicMax(mp + 1, __float_as_int(h4.y));
  atomicMax(mp + 2, __float_as_int(h4.z));
  atomicMax(mp + 3, __float_as_int(h4.w));
  if (q == 0) atomicAdd(&gcnt[g], 1.0f);
}

__global__ __launch_bounds__(256) void k_pool_final(const float* gsum, const float* gmax,
                                                    const float* gcnt, float* out, int g) {
  int i = blockIdx.x * 256 + threadIdx.x;
  if (i >= g * 2 * D) return;
  int gi = i / (2 * D);
  int f = i % (2 * D);
  float r;
  if (f < D)
    r = gsum[(size_t)gi * D + f] / fmaxf(gcnt[gi], 1.0f);
  else
    r = gmax[(size_t)gi * D + (f - D)];
  out[i] = r;
}

// ---------------- host side ----------------
extern "C" void kernel_launch(void* const* d_in, const int* in_sizes, int n_in,
                              void* d_out, int out_size, void* d_ws, size_t ws_size,
                              hipStream_t stream) {
  const float* x     = (const float*)d_in[0];
  const float* Wa    = (const float*)d_in[1];
  const float* ba    = (const float*)d_in[2];
  const float* ga    = (const float*)d_in[3];
  const float* be    = (const float*)d_in[4];
  const float* rm    = (const float*)d_in[5];
  const float* rv    = (const float*)d_in[6];
  const int*   ei    = (const int*)d_in[7];
  const int*   batch = (const int*)d_in[8];

  const int N = in_sizes[0] / D;
  const int E = in_sizes[7] / 2;
  const int G = out_size / (2 * D);
  const int* srcA = ei;
  const int* dstA = ei + E;

  // workspace layout (floats): 2 ping-pong feature buffers + dinv + pool scratch
  float* buf0 = (float*)d_ws;
  float* buf1 = buf0 + (size_t)N * D;
  float* dinv = buf1 + (size_t)N * D;
  float* gsum = dinv + N;
  float* gmax = gsum + (size_t)G * D;
  float* gcnt = gmax + (size_t)G * D;

  k_deg_init<<<(N + 255) / 256, 256, 0, stream>>>(dinv, N);
  k_deg_accum<<<(E + 255) / 256, 256, 0, stream>>>(dinv, dstA, E);
  k_dinv<<<(N + 255) / 256, 256, 0, stream>>>(dinv, N);

  const int rowblks = (N + 15) / 16;
  const float* hin = x;
  for (int l = 0; l < 3; ++l) {
    k_gemm_wmma<<<(rowblks + 7) / 8, 256, 0, stream>>>(hin, Wa + (size_t)l * D * D, buf0, N);
    k_agg_init<<<(N * 32 + 255) / 256, 256, 0, stream>>>(buf0, buf1, dinv, N);
    k_agg_edges<<<(E * 32 + 255) / 256, 256, 0, stream>>>(buf0, buf1, dinv, srcA, dstA, E);
    k_bn_relu<<<(N * D + 255) / 256, 256, 0, stream>>>(buf1, buf1, ba + l * D, ga + l * D,
                                                       be + l * D, rm + l * D, rv + l * D,
                                                       N * D);
    hin = buf1;
  }

  k_pool_zero<<<(G * D + 255) / 256, 256, 0, stream>>>(gsum, gmax, gcnt, G);
  k_pool_accum<<<(N * 32 + 255) / 256, 256, 0, stream>>>(buf1, batch, gsum, gmax, gcnt, N);
  k_pool_final<<<(G * 2 * D + 255) / 256, 256, 0, stream>>>(gsum, gmax, gcnt, (float*)d_out, G);
}